// LDRModel_10419590660372
// MI455X (gfx1250) — compile-verified
//
#include <hip/hip_runtime.h>
#include <math.h>

typedef float v2f __attribute__((ext_vector_type(2)));
typedef float v8f __attribute__((ext_vector_type(8)));

#define PI_F 3.14159265358979323846f

// ---------------------------------------------------------------------------
// f32 WMMA helper: D = A(16x4) * B(4x16) + C  on gfx1250
// ---------------------------------------------------------------------------
__device__ __forceinline__ v8f wmma4(v2f a, v2f b, v8f c) {
  return __builtin_amdgcn_wmma_f32_16x16x4_f32(
      /*neg_a=*/false, a, /*neg_b=*/false, b,
      /*c_mod=*/(short)0, c, /*reuse_a=*/false, /*reuse_b=*/false);
}

// ---------------------------------------------------------------------------
// Batched complex GEMM  D = A * B   (square SxS, S in {48,32}, zero-padded)
// A complex (strideA==0 -> shared constant, e.g. DFT matrix W)
// B complex (BCOMPLEX=true) or real (BCOMPLEX=false) -- specialized at
// compile time so the WMMA inner loop is branch-free.
// One wave (32 lanes) per 16x16 output tile; blockDim = (32, tiles*tiles).
// Complex product via 3 accumulators: Re = RR - II, Im = RI + IR.
// ---------------------------------------------------------------------------
template <bool BCOMPLEX>
__global__ void __launch_bounds__(288)
cgemm_kernel(const float* __restrict__ aRe, const float* __restrict__ aIm, long strideA,
             const float* __restrict__ bRe, const float* __restrict__ bIm, long strideB,
             float* __restrict__ dRe, float* __restrict__ dIm, long strideD,
             int S, int tiles) {
  const int batch = blockIdx.x;
  const int lane  = threadIdx.x;           // 0..31
  const int tile  = threadIdx.y;           // 0..tiles*tiles-1
  const int tm = (tile / tiles) * 16;
  const int tn = (tile % tiles) * 16;
  const int r = lane & 15;                 // row/col within tile
  const int h = lane >> 4;                 // half-wave select

  const float* are = aRe + (long)batch * strideA;
  const float* aim = aIm + (long)batch * strideA;
  const float* bre = bRe + (long)batch * strideB;
  const float* bim = BCOMPLEX ? (bIm + (long)batch * strideB) : nullptr;

  v8f accRR = {}, accII = {}, accIm = {};
  const int arow = tm + r;
  const int bcol = tn + r;

  for (int k0 = 0; k0 < S; k0 += 4) {
    const int ka = k0 + 2 * h;             // per-lane K offset (ISA 16x4 layout)
    // A fragments: consecutive floats at even offset -> 8B-aligned b64 loads
    v2f ar = *(const v2f*)(are + arow * S + ka);
    v2f ai = *(const v2f*)(aim + arow * S + ka);
    v2f br = { bre[ka * S + bcol], bre[(ka + 1) * S + bcol] };
    accRR = wmma4(ar, br, accRR);
    accIm = wmma4(ai, br, accIm);
    if (BCOMPLEX) {
      v2f bi = { bim[ka * S + bcol], bim[(ka + 1) * S + bcol] };
      accII = wmma4(ai, bi, accII);
      accIm = wmma4(ar, bi, accIm);
    }
  }

  float* dre = dRe + (long)batch * strideD;
  float* dim = dIm + (long)batch * strideD;
#pragma unroll
  for (int i = 0; i < 8; ++i) {
    const int row = tm + i + 8 * h;        // C/D layout: VGPR i -> rows i, i+8
    dre[row * S + bcol] = accRR[i] - accII[i];
    dim[row * S + bcol] = accIm[i];
  }
}

// ---------------------------------------------------------------------------
// DFT matrices, padded SxS with logical NxN (W is symmetric: fft2 = W X W)
// forward: e^{-2pi i jk/N}; inverse: conj/N
// ---------------------------------------------------------------------------
__global__ void dftmat_kernel(float* wfre, float* wfim, float* wire, float* wiim,
                              int S, int N) {
  int idx = blockIdx.x * blockDim.x + threadIdx.x;
  if (idx >= S * S) return;
  int j = idx / S, k = idx % S;
  float re = 0.f, im = 0.f, ir = 0.f, ii = 0.f;
  if (j < N && k < N) {
    float ang = -2.0f * PI_F * (float)((j * k) % N) / (float)N;
    re = __cosf(ang); im = __sinf(ang);
    ir = re / (float)N; ii = -im / (float)N;
  }
  wfre[idx] = re; wfim[idx] = im; wire[idx] = ir; wiim[idx] = ii;
}

// ---------------------------------------------------------------------------
// Periodized Gabor spatial arrays (33 of them), padded 48x48 logical 40x40.
// id 0: phi (sigma=3.2, theta=0, xi=0, slant=1)
// id 1+t / 9+t:  j=0 wavelet / envelope  (sigma=0.8,  xi=3pi/4, slant=0.5)
// id 17+t / 25+t: j=1 wavelet / envelope (sigma=1.6, xi=3pi/8, slant=0.5)
// ---------------------------------------------------------------------------
__global__ void gabor_kernel(float* gre, float* gim, int S, int N) {
  const int id = blockIdx.x;
  float sigma, theta, xi, slant;
  if (id == 0) { sigma = 3.2f; theta = 0.f; xi = 0.f; slant = 1.f; }
  else {
    int q = id - 1, grp = q / 8, t = q % 8;
    theta = (float)(3 - t) * (PI_F / 8.f);
    int j = grp >> 1;
    sigma = 0.8f * (j ? 2.f : 1.f);
    float xib = j ? (3.f * PI_F / 8.f) : (3.f * PI_F / 4.f);
    xi = ((grp & 1) == 0) ? xib : 0.f;
    slant = 0.5f;
  }
  const float ct = __cosf(theta), st = __sinf(theta);
  const float inv2s = 1.f / (2.f * sigma * sigma);
  const float c00   = (ct * ct + slant * slant * st * st) * inv2s;
  const float c11   = (st * st + slant * slant * ct * ct) * inv2s;
  const float c01x2 = 2.f * ct * st * (1.f - slant * slant) * inv2s;
  const float norm  = slant / (2.f * PI_F * sigma * sigma);
  for (int cell = threadIdx.x; cell < S * S; cell += blockDim.x) {
    int x = cell / S, y = cell % S;
    float re = 0.f, im = 0.f;
    if (x < N && y < N) {
#pragma unroll
      for (int ex = -2; ex <= 2; ++ex)
#pragma unroll
        for (int ey = -2; ey <= 2; ++ey) {
          float xg = (float)x + (float)(ex * N);
          float yg = (float)y + (float)(ey * N);
          float e  = __expf(-(c00 * xg * xg + c01x2 * xg * yg + c11 * yg * yg));
          float ph = xi * (xg * ct + yg * st);
          re += e * __cosf(ph);
          im += e * __sinf(ph);
        }
      re *= norm; im *= norm;
    }
    gre[(long)id * S * S + cell] = re;
    gim[(long)id * S * S + cell] = im;
  }
}

// Fourier-domain filters from the 33 spectra: morlet_hat = Re(WV) - Re(K*ENV),
// K = WV[0,0]/ENV[0,0] (zero-mean correction). phi0 = Re(spec[0]).
__global__ void combine_filters_kernel(const float* __restrict__ sre,
                                       const float* __restrict__ sim,
                                       int S, float* phi0, float* psi0, float* psi1) {
  int idx = blockIdx.x * blockDim.x + threadIdx.x;
  const int A = S * S;
  if (idx >= A) return;
  phi0[idx] = sre[idx];
  for (int j = 0; j < 2; ++j)
    for (int t = 0; t < 8; ++t) {
      int wv = 1 + j * 16 + t, env = 9 + j * 16 + t;
      float wr0 = sre[(long)wv * A],  wi0 = sim[(long)wv * A];
      float er0 = sre[(long)env * A], ei0 = sim[(long)env * A];
      float den = er0 * er0 + ei0 * ei0 + 1e-30f;
      float Kr = (wr0 * er0 + wi0 * ei0) / den;
      float Ki = (wi0 * er0 - wr0 * ei0) / den;
      float val = sre[(long)wv * A + idx]
                - (Kr * sre[(long)env * A + idx] - Ki * sim[(long)env * A + idx]);
      (j ? psi1 : psi0)[(long)t * A + idx] = val;
    }
}

// phi1 = periodize(phi0, 1): mean of the four 20x20 quadrants; 32-padded out.
__global__ void periodize_phi_kernel(const float* __restrict__ phi0, float* phi1) {
  int idx = blockIdx.x * blockDim.x + threadIdx.x;
  if (idx >= 32 * 32) return;
  int k1 = idx / 32, k2 = idx % 32;
  float v = 0.f;
  if (k1 < 20 && k2 < 20)
    v = 0.25f * (phi0[k1 * 48 + k2] + phi0[(k1 + 20) * 48 + k2] +
                 phi0[k1 * 48 + (k2 + 20)] + phi0[(k1 + 20) * 48 + (k2 + 20)]);
  phi1[idx] = v;
}

// Reflect-pad (32 -> 40) into zero-padded 48x48 real buffer.
__global__ void pad_kernel(const float* __restrict__ x, float* __restrict__ xp) {
  const int img = blockIdx.x;  // 0..383 = b*3+c
  for (int cell = threadIdx.x; cell < 48 * 48; cell += blockDim.x) {
    int i = cell / 48, j = cell % 48;
    float v = 0.f;
    if (i < 40 && j < 40) {
      int si = i - 4; si = si < 0 ? -si : (si > 31 ? 62 - si : si);
      int sj = j - 4; sj = sj < 0 ? -sj : (sj > 31 ? 62 - sj : sj);
      v = x[(long)img * 1024 + si * 32 + sj];
    }
    xp[(long)img * 2304 + cell] = v;
  }
}

// out[img*nfilt+t] = spec[img] * psi[t]   (pointwise, full 48x48)
__global__ void cmul_expand_kernel(const float* __restrict__ xre, const float* __restrict__ xim,
                                   const float* __restrict__ psi, int nfilt,
                                   float* __restrict__ ore, float* __restrict__ oim) {
  const int img = blockIdx.x, t = blockIdx.y;
  const long ib = (long)img * 2304;
  const long ob = ((long)img * nfilt + t) * 2304;
  const float* f = psi + (long)t * 2304;
  for (int c = threadIdx.x; c < 2304; c += blockDim.x) {
    float fv = f[c];
    ore[ob + c] = xre[ib + c] * fv;
    oim[ob + c] = xim[ib + c] * fv;
  }
}

// out20 = fold_2( spec40 * filt40 ): spectral alias-sum == spatial [::2,::2]
__global__ void mulfold_kernel(const float* __restrict__ xre, const float* __restrict__ xim,
                               const float* __restrict__ filt, int nfilt,
                               float* __restrict__ ore, float* __restrict__ oim) {
  const int img = blockIdx.x, t = blockIdx.y;
  const long ib = (long)img * 2304;
  const long ob = ((long)img * nfilt + t) * 1024;
  const float* f = filt + (long)t * 2304;
  for (int c = threadIdx.x; c < 1024; c += blockDim.x) {
    int k1 = c / 32, k2 = c % 32;
    float re = 0.f, im = 0.f;
    if (k1 < 20 && k2 < 20) {
#pragma unroll
      for (int a = 0; a < 2; ++a)
#pragma unroll
        for (int b = 0; b < 2; ++b) {
          int idx = (k1 + a * 20) * 48 + (k2 + b * 20);
          float fv = f[idx];
          re += xre[ib + idx] * fv;
          im += xim[ib + idx] * fv;
        }
      re *= 0.25f; im *= 0.25f;
    }
    ore[ob + c] = re;
    oim[ob + c] = im;
  }
}

// Modulus
__global__ void cabs_kernel(const float* __restrict__ re, const float* __restrict__ im,
                            float* __restrict__ out, long n) {
  long i = (long)blockIdx.x * blockDim.x + threadIdx.x;
  if (i < n) out[i] = sqrtf(re[i] * re[i] + im[i] * im[i]);
}

// Fused lowpass: fold(spec*phi) to 10x10, inverse 10-pt 2D DFT (real part),
// unpad 1-px border -> 8x8, scattered into f at feature slot s.
__global__ void __launch_bounds__(128)
lowpass10_kernel(const float* __restrict__ sre, const float* __restrict__ sim,
                 const float* __restrict__ filt, int S, int fold,
                 int nfilt, int s_base, int s_mul, float* __restrict__ fbuf) {
  __shared__ float zr[100], zi[100], ctab[10], stab[10];
  const int bi = blockIdx.x, tid = threadIdx.x;
  const long base = (long)bi * S * S;
  if (tid < 10) {
    float a = 2.f * PI_F * (float)tid / 10.f;
    ctab[tid] = __cosf(a); stab[tid] = __sinf(a);
  }
  if (tid < 100) {
    int k1 = tid / 10, k2 = tid % 10;
    float re = 0.f, im = 0.f;
    for (int a = 0; a < fold; ++a)
      for (int b = 0; b < fold; ++b) {
        int idx = (k1 + a * 10) * S + (k2 + b * 10);
        float fv = filt[idx];
        re += sre[base + idx] * fv;
        im += sim[base + idx] * fv;
      }
    float inv = 1.f / (float)(fold * fold);
    zr[tid] = re * inv; zi[tid] = im * inv;
  }
  __syncthreads();
  if (tid < 64) {
    int py = tid / 8 + 1, px = tid % 8 + 1;
    float acc = 0.f;
    for (int k1 = 0; k1 < 10; ++k1)
      for (int k2 = 0; k2 < 10; ++k2) {
        int m = (k1 * py + k2 * px) % 10;
        acc += zr[k1 * 10 + k2] * ctab[m] - zi[k1 * 10 + k2] * stab[m];
      }
    acc *= 0.01f;  // 1/100 inverse-DFT normalization
    int bc = bi / nfilt, t = bi % nfilt;
    int b = bc / 3, c = bc % 3;
    int s = s_base + t * s_mul;
    fbuf[(long)b * 15552 + ((long)(c * 81 + s) * 64) + (py - 1) * 8 + (px - 1)] = acc;
  }
}

// Training-mode batch norm over B=128 + ReLU, in place (thread per feature).
__global__ void bn_kernel(float* __restrict__ f, const float* __restrict__ gamma,
                          const float* __restrict__ beta) {
  int j = blockIdx.x * blockDim.x + threadIdx.x;
  if (j >= 15552) return;
  float mu = 0.f;
  for (int b = 0; b < 128; ++b) mu += f[(long)b * 15552 + j];
  mu *= (1.f / 128.f);
  float var = 0.f;
  for (int b = 0; b < 128; ++b) {
    float d = f[(long)b * 15552 + j] - mu;
    var += d * d;
  }
  var *= (1.f / 128.f);
  float inv = rsqrtf(var + 1e-5f) * gamma[j];
  float be = beta[j];
  for (int b = 0; b < 128; ++b) {
    float v = (f[(long)b * 15552 + j] - mu) * inv + be;
    f[(long)b * 15552 + j] = v > 0.f ? v : 0.f;
  }
}

// t[i,b,q] = sum_{s} h_i[s] * f[b, s+q], q<10  (only first-10 lags needed)
__global__ void __launch_bounds__(256)
tdot_kernel(const float* __restrict__ H, const float* __restrict__ f,
            float* __restrict__ tbuf) {
  const int blk = blockIdx.x;          // (i*128+b)*10 + q
  const int q = blk % 10, ib = blk / 10;
  const int b = ib % 128, i = ib / 128;
  const int n = 15552;
  const float* h  = H + (long)i * n;
  const float* fb = f + (long)b * n;
  float acc = 0.f;
  for (int s = threadIdx.x; s < n - q; s += 256) {
    if (s + 4096 < n) __builtin_prefetch(fb + s + 4096, 0, 1);  // global_prefetch_b8
    acc += h[s] * fb[s + q];
  }
  __shared__ float red[256];
  red[threadIdx.x] = acc;
  __syncthreads();
  for (int w = 128; w > 0; w >>= 1) {
    if (threadIdx.x < w) red[threadIdx.x] += red[threadIdx.x + w];
    __syncthreads();
  }
  if (threadIdx.x == 0) tbuf[blk] = red[0];
}

// y[b,j] = sum_i sum_{q<=j} g_i[j-q] * t[i,b,q]
__global__ void tfinal_kernel(const float* __restrict__ G, const float* __restrict__ tbuf,
                              float* __restrict__ y) {
  int idx = blockIdx.x * blockDim.x + threadIdx.x;
  if (idx >= 1280) return;
  int b = idx / 10, j = idx % 10;
  float acc = 0.f;
  for (int i = 0; i < 4; ++i)
    for (int q = 0; q <= j; ++q)
      acc += G[(long)i * 15552 + (j - q)] * tbuf[(i * 128 + b) * 10 + q];
  y[idx] = acc;
}

// ---------------------------------------------------------------------------
// Host-side orchestration
// ---------------------------------------------------------------------------
static inline void cgemm(hipStream_t st, int nbatch, int S,
                         const float* aRe, const float* aIm, long sA,
                         const float* bRe, const float* bIm, long sB,
                         float* dRe, float* dIm, long sD) {
  const int tps = S / 16;
  dim3 block(32, tps * tps);
  if (bIm)
    cgemm_kernel<true><<<nbatch, block, 0, st>>>(aRe, aIm, sA, bRe, bIm, sB,
                                                 dRe, dIm, sD, S, tps);
  else
    cgemm_kernel<false><<<nbatch, block, 0, st>>>(aRe, aIm, sA, bRe, nullptr, sB,
                                                  dRe, dIm, sD, S, tps);
}

extern "C" void kernel_launch(void* const* d_in, const int* in_sizes, int n_in,
                              void* d_out, int out_size, void* d_ws, size_t ws_size,
                              hipStream_t stream) {
  const float* x     = (const float*)d_in[0];   // (128,3,32,32)
  const float* gamma = (const float*)d_in[1];   // (15552)
  const float* beta  = (const float*)d_in[2];   // (15552)
  const float* G     = (const float*)d_in[3];   // (4,15552)
  const float* H     = (const float*)d_in[4];   // (4,15552)
  float* out = (float*)d_out;                   // (128,10)

  char* ws = (char*)d_ws;
  size_t off = 0;
  auto alloc = [&](size_t nfloat) -> float* {
    float* p = (float*)(ws + off);
    off += ((nfloat * sizeof(float) + 255) / 256) * 256;
    return p;
  };

  const int A48 = 48 * 48, A32 = 32 * 32;
  const long NB1 = 3072L * A48;   // 8-filter-expanded 40-res complex batch

  // DFT matrices
  float* w40f_re = alloc(A48); float* w40f_im = alloc(A48);
  float* w40i_re = alloc(A48); float* w40i_im = alloc(A48);
  float* w20f_re = alloc(A32); float* w20f_im = alloc(A32);
  float* w20i_re = alloc(A32); float* w20i_im = alloc(A32);
  // Fourier filters (real, padded)
  float* phi0 = alloc(A48); float* phi1 = alloc(A32);
  float* psi0 = alloc(8L * A48); float* psi1 = alloc(8L * A48);
  // Spectra / work buffers
  float* xf_re = alloc(384L * A48); float* xf_im = alloc(384L * A48);
  float* b1_re = alloc(NB1); float* b1_im = alloc(NB1);
  float* b2_re = alloc(NB1); float* b2_im = alloc(NB1);
  float* br    = alloc(NB1);                 // real buffer (|.| results)
  float* fbuf  = alloc(128L * 15552);        // scattering features
  float* tbuf  = alloc(4L * 128 * 10);       // truncated correlations

  // Aliased sub-buffers (lifetimes disjoint from their parents' other uses)
  float* a20_re = b2_re;               float* a20_im = b2_im;
  float* b20_re = b2_re + 3072L * A32; float* b20_im = b2_im + 3072L * A32;
  float* r20 = br;
  float* xp  = br;                          // padded input (before U1a exists)
  float* gsp_re = b1_re;  float* gsp_im = b1_im;           // 33 spatial gabors
  float* spc_re = b1_re + 33L * A48; float* spc_im = b1_im + 33L * A48;

  // ---- Filter bank construction (device, deterministic) ----
  dftmat_kernel<<<(A48 + 255) / 256, 256, 0, stream>>>(w40f_re, w40f_im, w40i_re, w40i_im, 48, 40);
  dftmat_kernel<<<(A32 + 255) / 256, 256, 0, stream>>>(w20f_re, w20f_im, w20i_re, w20i_im, 32, 20);
  gabor_kernel<<<33, 256, 0, stream>>>(gsp_re, gsp_im, 48, 40);
  cgemm(stream, 33, 48, w40f_re, w40f_im, 0, gsp_re, gsp_im, A48, b2_re, b2_im, A48);
  cgemm(stream, 33, 48, b2_re, b2_im, A48, w40f_re, w40f_im, 0, spc_re, spc_im, A48);
  combine_filters_kernel<<<(A48 + 255) / 256, 256, 0, stream>>>(spc_re, spc_im, 48, phi0, psi0, psi1);
  periodize_phi_kernel<<<(A32 + 255) / 256, 256, 0, stream>>>(phi0, phi1);

  // ---- Pad + forward DFT of input:  XF = W40 * xp * W40 ----
  pad_kernel<<<384, 256, 0, stream>>>(x, xp);
  cgemm(stream, 384, 48, w40f_re, w40f_im, 0, xp, nullptr, A48, b2_re, b2_im, A48);
  cgemm(stream, 384, 48, b2_re, b2_im, A48, w40f_re, w40f_im, 0, xf_re, xf_im, A48);

  // ---- Order 0: S0 = lowpass(x, phi0, k=4) ----
  lowpass10_kernel<<<384, 128, 0, stream>>>(xf_re, xf_im, phi0, 48, 4, 1, 0, 0, fbuf);

  // ---- Order 1, j1=0: U1a = |ifft(XF*psi0)|, UF1 = fft(U1a) ----
  cmul_expand_kernel<<<dim3(384, 8), 256, 0, stream>>>(xf_re, xf_im, psi0, 8, b1_re, b1_im);
  cgemm(stream, 3072, 48, w40i_re, w40i_im, 0, b1_re, b1_im, A48, b2_re, b2_im, A48);
  cgemm(stream, 3072, 48, b2_re, b2_im, A48, w40i_re, w40i_im, 0, b1_re, b1_im, A48);
  cabs_kernel<<<(int)((NB1 + 255) / 256), 256, 0, stream>>>(b1_re, b1_im, br, NB1);
  cgemm(stream, 3072, 48, w40f_re, w40f_im, 0, br, nullptr, A48, b2_re, b2_im, A48);
  cgemm(stream, 3072, 48, b2_re, b2_im, A48, w40f_re, w40f_im, 0, b1_re, b1_im, A48); // UF1
  lowpass10_kernel<<<3072, 128, 0, stream>>>(b1_re, b1_im, phi0, 48, 4, 8, 1, 1, fbuf);

  // ---- Order 1, j1=1: fold(XF*psi1) -> 20-res, |ifft20|, fft20, lowpass ----
  const long NB20 = 3072L * A32;
  mulfold_kernel<<<dim3(384, 8), 256, 0, stream>>>(xf_re, xf_im, psi1, 8, a20_re, a20_im);
  cgemm(stream, 3072, 32, w20i_re, w20i_im, 0, a20_re, a20_im, A32, b20_re, b20_im, A32);
  cgemm(stream, 3072, 32, b20_re, b20_im, A32, w20i_re, w20i_im, 0, a20_re, a20_im, A32);
  cabs_kernel<<<(int)((NB20 + 255) / 256), 256, 0, stream>>>(a20_re, a20_im, r20, NB20);
  cgemm(stream, 3072, 32, w20f_re, w20f_im, 0, r20, nullptr, A32, b20_re, b20_im, A32);
  cgemm(stream, 3072, 32, b20_re, b20_im, A32, w20f_re, w20f_im, 0, a20_re, a20_im, A32);
  lowpass10_kernel<<<3072, 128, 0, stream>>>(a20_re, a20_im, phi1, 32, 2, 8, 9, 1, fbuf);

  // ---- Order 2: for each t2, fold(UF1*psi1[t2]) -> |ifft20| -> lowpass ----
  for (int t2 = 0; t2 < 8; ++t2) {
    mulfold_kernel<<<dim3(3072, 1), 256, 0, stream>>>(b1_re, b1_im, psi1 + (long)t2 * A48, 1,
                                                      a20_re, a20_im);
    cgemm(stream, 3072, 32, w20i_re, w20i_im, 0, a20_re, a20_im, A32, b20_re, b20_im, A32);
    cgemm(stream, 3072, 32, b20_re, b20_im, A32, w20i_re, w20i_im, 0, a20_re, a20_im, A32);
    cabs_kernel<<<(int)((NB20 + 255) / 256), 256, 0, stream>>>(a20_re, a20_im, r20, NB20);
    cgemm(stream, 3072, 32, w20f_re, w20f_im, 0, r20, nullptr, A32, b20_re, b20_im, A32);
    cgemm(stream, 3072, 32, b20_re, b20_im, A32, w20f_re, w20f_im, 0, a20_re, a20_im, A32);
    lowpass10_kernel<<<3072, 128, 0, stream>>>(a20_re, a20_im, phi1, 32, 2, 8, 17 + t2, 8, fbuf);
  }

  // ---- BN + ReLU, truncated Toeplitz, output ----
  bn_kernel<<<(15552 + 255) / 256, 256, 0, stream>>>(fbuf, gamma, beta);
  tdot_kernel<<<4 * 128 * 10, 256, 0, stream>>>(H, fbuf, tbuf);
  tfinal_kernel<<<(1280 + 255) / 256, 256, 0, stream>>>(G, tbuf, out);
}